// MyAttacker_73607149519451
// MI455X (gfx1250) — compile-verified
//
#include <hip/hip_runtime.h>
#include <stdint.h>
#include <math.h>

// ---------------------------------------------------------------------------
// Nettack Thm 5.1 dense [P,N] update, MI455X (gfx1250).
// Bandwidth-bound: ~123 MB traffic vs 23.3 TB/s => ~5.5us floor. No GEMM
// structure => WMMA inapplicable; CDNA5-specific paths used instead:
//   * global_load_async_to_lds_b128 staging of the two per-p adjacency rows
//     (ASYNCcnt-tracked async copy path) + s_wait_asynccnt
//   * nontemporal 128-bit stores for the 82MB streaming output (preserve L2
//     for adjacency-row reuse; L2 = 192MB)
//   * v_rcp/v_rsq/v_sqrt hardware transcendentals (TRANS32, co-issue with
//     VALU) instead of IEEE div expansions: keeps VALU below the HBM floor.
// ---------------------------------------------------------------------------

#define TB 256
#define TV 4096   // v-tile bytes per block (N=10000 = 2*4096 + 1808, all %16==0)

typedef float f32x4 __attribute__((ext_vector_type(4)));
typedef int   i32x4 __attribute__((ext_vector_type(4)));
// Exact pointee type of the gfx1250 async-b128 builtin params (gcc vector_size).
typedef int   v4i   __attribute__((vector_size(16)));

#if defined(__AMDGCN__)
#if __has_builtin(__builtin_amdgcn_global_load_async_to_lds_b128) && \
    __has_builtin(__builtin_amdgcn_s_wait_asynccnt)
#define USE_ASYNC 1
#else
#define USE_ASYNC 0
#warning "gfx1250 device pass: async-to-LDS builtins NOT found; sync LDS fallback in use"
#endif
#else
#define USE_ASYNC 0
#endif

__device__ __forceinline__ void stage16(const uint8_t* g, uint8_t* l) {
#if USE_ASYNC
  // Per-lane 16B async copy, global -> LDS. Low 32 bits of a flat shared
  // pointer are the LDS byte offset (ISA 10.2), AS1 ptr is the full VA.
  __builtin_amdgcn_global_load_async_to_lds_b128(
      (__attribute__((address_space(1))) v4i*)(uintptr_t)g,
      (__attribute__((address_space(3))) v4i*)(uint32_t)(uintptr_t)l,
      0, 0);
#else
  *(i32x4*)l = *(const i32x4*)g;
#endif
}

// Hardware transcendentals: operands are small positive integers (node
// degrees), so the ~1ulp v_rcp/v_rsq/v_sqrt are safe; avoids the ~10-instr
// IEEE div/rsqrt fixup expansions.
__device__ __forceinline__ float fast_rcp(float x) {
#if defined(__AMDGCN__)
  return __builtin_amdgcn_rcpf(x);
#else
  return 1.0f / x;
#endif
}
__device__ __forceinline__ float fast_rsq(float x) {
#if defined(__AMDGCN__)
  return __builtin_amdgcn_rsqf(x);
#else
  return 1.0f / sqrtf(x);
#endif
}
__device__ __forceinline__ float fast_sqrt(float x) {
#if defined(__AMDGCN__)
  return __builtin_amdgcn_sqrtf(x);
#else
  return sqrtf(x);
#endif
}

__global__ __launch_bounds__(TB) void nettack_ahat2_kernel(
    const uint8_t* __restrict__ adj,   // [N,N] bool (symmetric)
    const int*     __restrict__ degs,  // [N]
    const float*   __restrict__ vals,  // [N]
    const int*     __restrict__ pe,    // [P,2]
    const int*     __restrict__ uPtr,  // [1]
    float*         __restrict__ out,   // [P,N]
    int N)
{
  __shared__ __align__(16) uint8_t sM[TV];
  __shared__ __align__(16) uint8_t sN[TV];

  const int tid = threadIdx.x;
  const int p   = blockIdx.y;
  const int v0  = blockIdx.x * TV;
  const int rem = (N - v0 < TV) ? (N - v0) : TV;   // multiple of 16 for N=10000

  const int u = uPtr[0];
  const int m = pe[2 * p];
  const int n = pe[2 * p + 1];

  const uint8_t* rowM = adj + (size_t)m * N;
  const uint8_t* rowN = adj + (size_t)n * N;
  const uint8_t* rowU = adj + (size_t)u * N;

  // ---- async-stage the two per-p adjacency row tiles into LDS -------------
  {
    const int byteOff = tid * 16;
    if (byteOff < rem) {
      stage16(rowM + v0 + byteOff, sM + byteOff);
      stage16(rowN + v0 + byteOff, sN + byteOff);
    }
  }
  if (tid == 0) __builtin_prefetch(rowU + v0, 0, 3);  // global_prefetch_b8

  // ---- per-p scalars (uniform; broadcast loads, L2-hot) -------------------
  const bool  exists     = rowM[n] != 0;            // adj[m,n]
  const float delta      = exists ? -1.0f : 1.0f;
  const bool  insb       = !exists;                 // ins = (delta > 0)
  const bool  u_in_edge  = (u == m) | (u == n);
  const float degs_u     = (float)degs[u];
  const float degs_m     = (float)degs[m];
  const float degs_n     = (float)degs[n];
  const float degs_new_u = degs_u + (u_in_edge ? delta : 0.0f);
  const float degs_new_m = degs_m + delta;
  const float degs_new_n = degs_n + delta;
  const bool  a_um       = rowU[m] != 0;
  const bool  a_un       = rowU[n] != 0;
  const bool  a_un_after = (u == m) ? insb : a_un;
  const bool  a_um_after = (u == n) ? insb : a_um;

  const float inv_degs_u     = fast_rcp(degs_u);
  const float inv_degs_new_u = fast_rcp(degs_new_u);
  const float rsq_new_u      = fast_rsq(degs_new_u);
  const float sq_degs_u      = fast_sqrt(degs_u);
  // p-uniform select payloads (SGPR operands of v_cndmask in the hot loop)
  const float cM_old = a_um       ? fast_rcp(degs_m)     : 0.0f;
  const float cM_new = a_um_after ? fast_rcp(degs_new_m) : 0.0f;
  const float cN_old = a_un       ? fast_rcp(degs_n)     : 0.0f;
  const float cN_new = a_un_after ? fast_rcp(degs_new_n) : 0.0f;

#if USE_ASYNC
  __builtin_amdgcn_s_wait_asynccnt(0);
#endif
  __syncthreads();

  // ---- streaming compute: 4 consecutive v per thread per step -------------
  const uint32_t* sM32 = (const uint32_t*)sM;
  const uint32_t* sN32 = (const uint32_t*)sN;
  const int dwords = rem >> 2;
  float* outRow = out + (size_t)p * N;

  for (int idx = tid; idx < dwords; idx += TB) {
    const int v = v0 + (idx << 2);
    const uint32_t mPack = sM32[idx];                        // LDS, stride-1 dwords
    const uint32_t nPack = sN32[idx];
    const uint32_t uPack = *(const uint32_t*)(rowU + v);     // L2-hot shared row
    const i32x4 dg = *(const i32x4*)(degs + v);
    const f32x4 vl = *(const f32x4*)(vals + v);

    f32x4 res;
#pragma unroll
    for (int k = 0; k < 4; ++k) {
      const int  vv   = v + k;
      const bool a_vm = ((mPack >> (8 * k)) & 0xFFu) != 0u;  // adj[v,m]=adj[m,v]
      const bool a_vn = ((nPack >> (8 * k)) & 0xFFu) != 0u;  // adj[v,n]=adj[n,v]
      const bool a_uv = ((uPack >> (8 * k)) & 0xFFu) != 0u;  // adj[u,v]

      const float degs_v     = (float)dg[k];
      const bool  in_edge_v  = (vv == m) | (vv == n);
      const float degs_new_v = in_edge_v ? (degs_v + delta) : degs_v;

      const bool veq_u         = (vv == u);
      const bool a_uv_sl       = a_uv | veq_u;
      const bool cond          = in_edge_v & u_in_edge & (!veq_u);
      const bool a_uv_after    = cond ? insb : a_uv;
      const bool a_uv_after_sl = a_uv_after | veq_u;
      const bool a_vn_after    = (vv == m) ? insb : a_vn;
      const bool a_vm_after    = (vv == n) ? insb : a_vm;

      const float rcp_v  = fast_rcp(degs_v);       // v_rcp_f32 (TRANS32)
      const float rcp_nv = fast_rcp(degs_new_v);
      const float mult   = rsq_new_u * fast_rsq(degs_new_v);

      const float s1 = sq_degs_u * fast_sqrt(degs_v) * vl[k]
                     - (a_uv_sl ? inv_degs_u : 0.0f)
                     - (a_uv    ? rcp_v      : 0.0f);
      const float s2 = (a_uv_after    ? rcp_nv         : 0.0f)
                     + (a_uv_after_sl ? inv_degs_new_u : 0.0f);
      const float s3 = (a_vm_after ? cM_new : 0.0f)
                     - (a_vm      ? cM_old : 0.0f);
      const float s4 = (a_vn_after ? cN_new : 0.0f)
                     - (a_vn      ? cN_old : 0.0f);

      res[k] = mult * ((s1 + s2) + (s3 + s4));
    }
    // 82MB streaming output: NT store, keep L2 for adjacency-row reuse.
    __builtin_nontemporal_store(res, (f32x4*)(outRow + v));
  }
}

extern "C" void kernel_launch(void* const* d_in, const int* in_sizes, int n_in,
                              void* d_out, int out_size, void* d_ws, size_t ws_size,
                              hipStream_t stream) {
  const uint8_t* adj  = (const uint8_t*)d_in[0];  // bool -> 1 byte/elt
  const int*     degs = (const int*)d_in[1];
  const float*   vals = (const float*)d_in[2];
  const int*     pe   = (const int*)d_in[3];
  const int*     uPtr = (const int*)d_in[4];
  float*         out  = (float*)d_out;

  const int N = in_sizes[1];       // 10000
  const int P = in_sizes[3] / 2;   // 2048

  dim3 grid((N + TV - 1) / TV, P); // (3, 2048)
  nettack_ahat2_kernel<<<grid, dim3(TB), 0, stream>>>(adj, degs, vals, pe, uPtr, out, N);
}